// DummyAttention_31379031065274
// MI455X (gfx1250) — compile-verified
//
#include <hip/hip_runtime.h>
#include <math.h>

typedef __attribute__((ext_vector_type(2))) float v2f;
typedef __attribute__((ext_vector_type(8))) float v8f;

constexpr int Bc = 2, Qc = 512, HIDc = 4096;
constexpr int HKVc = 8, Dc = 128, Gc = 4;
constexpr int Sc = 16384, Lc = 4096;
constexpr float SCALEc = 0.08838834764831845f; // 1/sqrt(128)

__device__ __forceinline__ v8f wmma_f32(v2f a, v2f b, v8f c) {
  // D = A(16x4) x B(4x16) + C(16x16), fp32
  return __builtin_amdgcn_wmma_f32_16x16x4_f32(false, a, false, b, (short)0, c,
                                               false, false);
}

// Async 16B/lane copy global -> LDS (CDNA5, tracked by ASYNCcnt).
__device__ __forceinline__ void async_copy16(unsigned int lds_byte,
                                             unsigned long long gaddr) {
  asm volatile("global_load_async_to_lds_b128 %0, %1, off"
               :
               : "v"(lds_byte), "v"(gaddr)
               : "memory");
}
__device__ __forceinline__ void wait_async_le9() {
  asm volatile("s_wait_asynccnt 0x9" ::: "memory");
}
__device__ __forceinline__ void wait_async_0() {
  asm volatile("s_wait_asynccnt 0x0" ::: "memory");
}
// Low 32 bits of a flat shared pointer == LDS byte offset.
__device__ __forceinline__ unsigned int lds_off(const void* p) {
  return (unsigned int)(unsigned long long)(uintptr_t)p;
}

// ---------------------------------------------------------------------------
// C[M,N] = A[M,K] @ W[N,K]^T (row-major), fp32 WMMA, async-LDS double buffer.
// Block = 256 thr = 8 waves; block tile 32(M) x 256(N); K-tile = 32.
// ---------------------------------------------------------------------------
constexpr int KT = 32;                 // K tile
constexpr int RS = 36;                 // padded LDS row stride (floats)
constexpr int WOFF = 32 * RS;          // W region offset within a buffer
constexpr int BUFF = (32 + 256) * RS;  // floats per buffer

__global__ __launch_bounds__(256) void gemm_awt(const float* __restrict__ A,
                                                const float* __restrict__ W,
                                                float* __restrict__ C, int M,
                                                int N, int K) {
  __shared__ float lds[2 * BUFF];
  const int tid = threadIdx.x;
  const int lane = tid & 31;
  const int w = tid >> 5;
  const int wm = w >> 2, wn = w & 3;
  const int h = lane >> 4, n16 = lane & 15;
  const int Mbase = blockIdx.y * 32;
  const int Nbase = blockIdx.x * 256;

  // ---- async staging addresses (per lane: one 16B chunk of a row) ----
  const int c16 = lane & 7;            // chunk within a 128B row
  const int arow = 4 * w + (lane >> 3);        // A local row 0..31
  const int wrow0 = 32 * w + (lane >> 3);      // first W local row for inst 0
  const unsigned long long Kb = (unsigned long long)K * 4ull;
  const unsigned long long agp =
      (unsigned long long)(uintptr_t)A + (unsigned long long)(Mbase + arow) * Kb +
      (unsigned long long)c16 * 16ull;
  const unsigned long long wgp0 =
      (unsigned long long)(uintptr_t)W + (unsigned long long)(Nbase + wrow0) * Kb +
      (unsigned long long)c16 * 16ull;
  const unsigned int lbase = lds_off(&lds[0]);
  const unsigned int a_lds = lbase + (unsigned int)((arow * RS + c16 * 4) * 4);
  const unsigned int w_lds =
      lbase + (unsigned int)((WOFF + wrow0 * RS + c16 * 4) * 4);

  auto issue = [&](int kt, int buf) {
    const unsigned int boff = (unsigned int)buf * (BUFF * 4);
    const unsigned long long go = (unsigned long long)kt * (KT * 4);
    async_copy16(a_lds + boff, agp + go);
#pragma unroll
    for (int i = 0; i < 8; ++i)
      async_copy16(w_lds + boff + (unsigned int)(4 * i * RS * 4),
                   wgp0 + go + (unsigned long long)(4 * i) * Kb);
  };

  v8f acc0 = {}, acc1 = {}, acc2 = {}, acc3 = {};
  const int NT = K / KT;
  issue(0, 0);
  for (int kt = 0; kt < NT; ++kt) {
    const int buf = kt & 1;
    if (kt + 1 < NT) {
      issue(kt + 1, buf ^ 1);
      wait_async_le9(); // oldest 9 (current tile) complete; next tile in flight
    } else {
      wait_async_0();
    }
    __syncthreads(); // all waves' staging visible
    const float* Ab = &lds[buf * BUFF];
    const float* Wb = &lds[buf * BUFF + WOFF];
#pragma unroll
    for (int kb = 0; kb < KT; kb += 4) {
      const int ko = kb + 2 * h;
      v2f a = *(const v2f*)&Ab[(wm * 16 + n16) * RS + ko];
      v2f b0 = *(const v2f*)&Wb[(wn * 64 + 0 + n16) * RS + ko];
      v2f b1 = *(const v2f*)&Wb[(wn * 64 + 16 + n16) * RS + ko];
      v2f b2 = *(const v2f*)&Wb[(wn * 64 + 32 + n16) * RS + ko];
      v2f b3 = *(const v2f*)&Wb[(wn * 64 + 48 + n16) * RS + ko];
      acc0 = wmma_f32(a, b0, acc0);
      acc1 = wmma_f32(a, b1, acc1);
      acc2 = wmma_f32(a, b2, acc2);
      acc3 = wmma_f32(a, b3, acc3);
    }
    __syncthreads(); // all reads done before buffer is re-issued
  }

  const int m0 = Mbase + wm * 16 + 8 * h;
#pragma unroll
  for (int j = 0; j < 8; ++j) {
    float* Crow = C + (size_t)(m0 + j) * N + Nbase + wn * 64 + n16;
    Crow[0] = acc0[j];
    Crow[16] = acc1[j];
    Crow[32] = acc2[j];
    Crow[48] = acc3[j];
  }
}

// ---------------------------------------------------------------------------
// inv_map: which (b*Q+q) token overwrote cache slot s (else -1)
// ---------------------------------------------------------------------------
__global__ void init_inv(int* __restrict__ inv) {
  int i = blockIdx.x * blockDim.x + threadIdx.x;
  if (i < Sc) inv[i] = -1;
}

__global__ void scat_inv(const int* __restrict__ lut, int* __restrict__ inv,
                         int n) {
  int i = blockIdx.x * blockDim.x + threadIdx.x;
  if (i < n) inv[lut[i]] = i;
}

// ---------------------------------------------------------------------------
// RoPE on K and V. Layout: (B*Q, HKV*D) row-major. rope_cache: (MAXPOS, 2, D)
// ---------------------------------------------------------------------------
__global__ __launch_bounds__(256) void rope_kv(
    const float* __restrict__ kin, const float* __restrict__ vin,
    const float* __restrict__ rope_cache, const int* __restrict__ pos_offs,
    float* __restrict__ kout, float* __restrict__ vout) {
  const int row = blockIdx.x; // b*Q + q
  const int b = row / Qc;
  const int q = row - b * Qc;
  const int pos = pos_offs[b] + q;
  const float* cosp = rope_cache + (size_t)pos * 2 * Dc;
  const float* sinp = cosp + Dc;
  const size_t rb = (size_t)row * (HKVc * Dc);
  const float* kr = kin + rb;
  const float* vr = vin + rb;
  float* ko = kout + rb;
  float* vo = vout + rb;
  for (int c = threadIdx.x; c < HKVc * Dc; c += blockDim.x) {
    const int d = c & (Dc - 1);
    const int base = c - d;
    const float cs = cosp[d], sn = sinp[d];
    float krot, vrot;
    if (d < Dc / 2) {
      krot = -kr[base + d + Dc / 2];
      vrot = -vr[base + d + Dc / 2];
    } else {
      krot = kr[base + d - Dc / 2];
      vrot = vr[base + d - Dc / 2];
    }
    ko[c] = kr[c] * cs + krot * sn;
    vo[c] = vr[c] * cs + vrot * sn;
  }
}

// ---------------------------------------------------------------------------
// Flash attention (fp32 WMMA). Grid: (Q/16, HKV, B); block = 128 thr = 4 waves
// (one per GQA query head sharing this KV head). K/V tiles gathered to LDS
// via async-to-LDS copies.
// ---------------------------------------------------------------------------
__global__ __launch_bounds__(128) void attn_fa(
    const float* __restrict__ qbuf, const float* __restrict__ krope,
    const float* __restrict__ vrope, const float* __restrict__ kv_cache,
    const int* __restrict__ kv_lut, const int* __restrict__ inv_map,
    const int* __restrict__ pos_offs, float* __restrict__ attn_out) {
  constexpr int KS = 132; // padded LDS row stride (floats) for K/V tiles
  constexpr int PS = 17;  // padded LDS row stride for P tiles
  __shared__ float k_lds[16 * KS];
  __shared__ float v_lds[16 * KS];
  __shared__ float p_lds[4 * 16 * PS];

  const int q0 = blockIdx.x * 16;
  const int kv = blockIdx.y;
  const int b = blockIdx.z;
  const int tid = threadIdx.x;
  const int lane = tid & 31;
  const int w = tid >> 5; // g index (query head within group)
  const int h = lane >> 4;
  const int n16 = lane & 15;
  const int hq = kv * Gc + w;
  const int pos_off = pos_offs[b];

  // Preload this wave's 16x128 Q tile into A-layout register fragments.
  v2f qf[32];
  const float* Qrow =
      qbuf + ((size_t)(b * Qc + q0 + n16) * HIDc + (size_t)hq * Dc);
#pragma unroll
  for (int s = 0; s < 32; ++s) qf[s] = *(const v2f*)(Qrow + 4 * s + 2 * h);

  v8f acc[8];
#pragma unroll
  for (int t = 0; t < 8; ++t) acc[t] = v8f{};
  float m_run[8], l_run[8];
#pragma unroll
  for (int j = 0; j < 8; ++j) {
    m_run[j] = -1.0e30f;
    l_run[j] = 0.0f;
  }

  const int lmax = pos_off + q0 + 15; // max causally-needed L index (< Lc)
  const int ntiles = (lmax >> 4) + 1;
  float* my_p = p_lds + w * 16 * PS;
  const unsigned int k_lds0 = lds_off(&k_lds[0]);
  const unsigned int v_lds0 = lds_off(&v_lds[0]);

  for (int t = 0; t < ntiles; ++t) {
    const int l0 = t * 16;
    __syncthreads(); // previous tile's LDS reads complete
    {
      // Cooperative gather of K/V 16x128 tiles: 8 threads/row, 16 floats each,
      // via CDNA5 async global->LDS copies (per-lane indirect addresses).
      const int row = tid >> 3;
      const int cb = (tid & 7) * 16;
      const int slot = kv_lut[b * Lc + l0 + row];
      const int im = inv_map[slot];
      const float* ksrc =
          (im >= 0) ? (krope + ((size_t)im * HKVc + kv) * Dc)
                    : (kv_cache + (((size_t)slot * 2 + 0) * HKVc + kv) * Dc);
      const float* vsrc =
          (im >= 0) ? (vrope + ((size_t)im * HKVc + kv) * Dc)
                    : (kv_cache + (((size_t)slot * 2 + 1) * HKVc + kv) * Dc);
      const unsigned int loff = (unsigned int)((row * KS + cb) * 4);
#pragma unroll
      for (int i = 0; i < 4; ++i) {
        async_copy16(k_lds0 + loff + 16u * i,
                     (unsigned long long)(uintptr_t)(ksrc + cb + 4 * i));
        async_copy16(v_lds0 + loff + 16u * i,
                     (unsigned long long)(uintptr_t)(vsrc + cb + 4 * i));
      }
      wait_async_0();
    }
    __syncthreads(); // K/V visible

    // Scores: S(16x16) = Qtile(16x128) @ Ktile^T, K-dim = 128 in steps of 4.
    v8f s = {};
#pragma unroll
    for (int st = 0; st < 32; ++st) {
      v2f bf = *(const v2f*)&k_lds[n16 * KS + 4 * st + 2 * h];
      s = wmma_f32(qf[st], bf, s);
    }

    // Causal mask + online softmax (row = j + 8*h, col = n16 within tile).
    float p[8], fsc[8];
#pragma unroll
    for (int j = 0; j < 8; ++j) {
      const int qrow = q0 + j + 8 * h; // global query index
      const int lcol = l0 + n16;       // global key index
      float sv = (lcol <= pos_off + qrow) ? s[j] * SCALEc : -1.0e30f;
      float rm = sv;
      rm = fmaxf(rm, __shfl_xor(rm, 1, 16));
      rm = fmaxf(rm, __shfl_xor(rm, 2, 16));
      rm = fmaxf(rm, __shfl_xor(rm, 4, 16));
      rm = fmaxf(rm, __shfl_xor(rm, 8, 16));
      const float mnew = fmaxf(m_run[j], rm);
      const float f = __expf(m_run[j] - mnew);
      const float pv = __expf(sv - mnew);
      float rs = pv;
      rs += __shfl_xor(rs, 1, 16);
      rs += __shfl_xor(rs, 2, 16);
      rs += __shfl_xor(rs, 4, 16);
      rs += __shfl_xor(rs, 8, 16);
      l_run[j] = l_run[j] * f + rs;
      m_run[j] = mnew;
      p[j] = pv;
      fsc[j] = f;
    }
#pragma unroll
    for (int tt = 0; tt < 8; ++tt)
#pragma unroll
      for (int j = 0; j < 8; ++j) acc[tt][j] *= fsc[j];

    // Reshape P: C-layout regs -> LDS -> A-layout fragments.
#pragma unroll
    for (int j = 0; j < 8; ++j) my_p[(j + 8 * h) * PS + n16] = p[j];
    __syncthreads(); // P visible (also keeps waves in lockstep for K/V reuse)

    // O(16x128) += P(16x16) @ Vtile(16x128)
#pragma unroll
    for (int lk = 0; lk < 16; lk += 4) {
      v2f pa = {my_p[n16 * PS + lk + 2 * h], my_p[n16 * PS + lk + 1 + 2 * h]};
#pragma unroll
      for (int tt = 0; tt < 8; ++tt) {
        v2f vb = {v_lds[(lk + 2 * h) * KS + tt * 16 + n16],
                  v_lds[(lk + 1 + 2 * h) * KS + tt * 16 + n16]};
        acc[tt] = wmma_f32(pa, vb, acc[tt]);
      }
    }
  }

  // Normalize and write out as (b, q, HID) with HID col = hq*D + d.
#pragma unroll
  for (int j = 0; j < 8; ++j) {
    const int qrow = q0 + j + 8 * h;
    const float invl = 1.0f / l_run[j];
    float* orow =
        attn_out + ((size_t)(b * Qc + qrow) * HIDc + (size_t)hq * Dc + n16);
#pragma unroll
    for (int tt = 0; tt < 8; ++tt) orow[tt * 16] = acc[tt][j] * invl;
  }
}

// ---------------------------------------------------------------------------
extern "C" void kernel_launch(void* const* d_in, const int* in_sizes, int n_in,
                              void* d_out, int out_size, void* d_ws,
                              size_t ws_size, hipStream_t stream) {
  const float* hs = (const float*)d_in[0];   // (B,Q,HID)
  const float* kvc = (const float*)d_in[1];  // (S,2,HKV,D)
  const float* rope = (const float*)d_in[2]; // (MAXPOS,2,D)
  const float* qw = (const float*)d_in[3];   // (HID,HID)
  const float* kw = (const float*)d_in[4];   // (HKV*D,HID)
  const float* vw = (const float*)d_in[5];   // (HKV*D,HID)
  const float* ow = (const float*)d_in[6];   // (HID,HID)
  const int* poff = (const int*)d_in[7];     // (B,)
  const int* dlut = (const int*)d_in[8];     // (B,Q)
  const int* klut = (const int*)d_in[9];     // (B,L)
  float* out = (float*)d_out;

  const int M = Bc * Qc;     // 1024
  const int KVD = HKVc * Dc; // 1024

  float* ws = (float*)d_ws;
  float* qb = ws;                           // M*HID
  float* kb = qb + (size_t)M * HIDc;        // M*KVD
  float* vb = kb + (size_t)M * KVD;         // M*KVD
  float* kr = vb + (size_t)M * KVD;         // M*KVD (roped K)
  float* vr = kr + (size_t)M * KVD;         // M*KVD (roped V)
  float* at = vr + (size_t)M * KVD;         // M*HID (attn pre-Oproj)
  int* inv = (int*)(at + (size_t)M * HIDc); // S ints

  // Q/K/V projections
  gemm_awt<<<dim3(HIDc / 256, M / 32), 256, 0, stream>>>(hs, qw, qb, M, HIDc,
                                                         HIDc);
  gemm_awt<<<dim3(KVD / 256, M / 32), 256, 0, stream>>>(hs, kw, kb, M, KVD,
                                                        HIDc);
  gemm_awt<<<dim3(KVD / 256, M / 32), 256, 0, stream>>>(hs, vw, vb, M, KVD,
                                                        HIDc);
  // Cache-slot inverse map
  init_inv<<<Sc / 256, 256, 0, stream>>>(inv);
  scat_inv<<<M / 256, 256, 0, stream>>>(dlut, inv, M);
  // RoPE on K and V
  rope_kv<<<M, 256, 0, stream>>>(kb, vb, rope, poff, kr, vr);
  // Flash attention over gathered KV
  attn_fa<<<dim3(Qc / 16, HKVc, Bc), 128, 0, stream>>>(qb, kr, vr, kvc, klut,
                                                       inv, poff, at);
  // Output projection
  gemm_awt<<<dim3(HIDc / 256, M / 32), 256, 0, stream>>>(at, ow, out, M, HIDc,
                                                         HIDc);
}